// AnisotropicDistance_59803124629518
// MI455X (gfx1250) — compile-verified
//
#include <hip/hip_runtime.h>

typedef float v2f __attribute__((ext_vector_type(2)));
typedef float v4f __attribute__((ext_vector_type(4)));
typedef float v8f __attribute__((ext_vector_type(8)));

#define JT 16  // j-tiles (of 16 cols) handled per wave

// out[b,i,j] = alpha_i * max(||p_i||^2 + ||p_j||^2 - 2 p_i.p_j + (||t_i||^2-2)*(s_i - p_j.t_i)^2, 0)
//            + beta_i * (s_i - p_j.t_i)^2,   s_i = p_i.t_i
//
// Tile scheme (orientation: D[m=j][n=i]):
//   A (16x4, M=j rows):  A[j][0..2] = p_j,  A[j][3] = ||p_j||^2   (shared by both WMMAs)
//   Bg (4x16, N=i cols): Bg[0..2][i] = p_i, Bg[3][i] = -0.5  -> Dg = p_j.p_i - 0.5||p_j||^2
//   Bx (4x16, N=i cols): Bx[0..2][i] = t_i, Bx[3][i] = 0     -> Dx = p_j.t_i
// so sq_dist = ||p_i||^2 - 2*Dg.
__global__ void __launch_bounds__(256)
aniso_wmma_kernel(const float* __restrict__ points,
                  const float* __restrict__ pdir,
                  const float* __restrict__ lin,
                  float* __restrict__ out,
                  int B, int N, int jBlocks) {
  const int lane = threadIdx.x & 31;
  const int wave = (int)((blockIdx.x * blockDim.x + threadIdx.x) >> 5);
  const int itCount = N >> 4;
  const int wavesPerB = itCount * jBlocks;
  const int b = wave / wavesPerB;
  if (b >= B) return;                       // wave-uniform (EXEC stays all-ones)
  const int rem = wave - b * wavesPerB;
  const int it = rem / jBlocks;
  const int jb = rem - it * jBlocks;
  const int i0 = it << 4;
  const int j0base = jb * (16 * JT);

  const int l16 = lane & 15;
  const int h = lane >> 4;                  // half-wave: K group (0 -> K=0,1 ; 1 -> K=2,3)

  // ---- per-lane i-column data (reused across all j-tiles of this wave) ----
  const int i = i0 + l16;
  const float* pI = points + ((size_t)b * N + i) * 3;
  const float* tI = pdir   + ((size_t)b * N + i) * 3;
  const float pix = pI[0], piy = pI[1], piz = pI[2];
  const float tix = tI[0], tiy = tI[1], tiz = tI[2];
  const float li  = lin[(size_t)b * N + i];
  const float s_i   = pix * tix + piy * tiy + piz * tiz;
  const float sq_i  = pix * pix + piy * piy + piz * piz;
  const float tn2m2 = tix * tix + tiy * tiy + tiz * tiz - 2.0f;
  const float alpha = 2.0f * (1.0f + li);
  const float beta  = 0.5f * (1.0f - li);

  // B operands: lane l reg v holds K = v + 2h, col N = l16
  v2f Bg, Bx;
  Bg.x = h ? piz   : pix;
  Bg.y = h ? -0.5f : piy;
  Bx.x = h ? tiz   : tix;
  Bx.y = h ? 0.0f  : tiy;

  const float* pBase = points + (size_t)b * N * 3;
  // lane's output row pointer (fixed i); its 8 values per tile are j0+8h .. j0+8h+7
  float* outRow = out + ((size_t)b * N + i) * (size_t)N;

  for (int jt = 0; jt < JT; ++jt) {
    const int j0 = j0base + (jt << 4);
    const int j = j0 + l16;
    const float* pJ = pBase + (size_t)j * 3;
    const float pjx = pJ[0], pjy = pJ[1], pjz = pJ[2];
    const float sq_j = pjx * pjx + pjy * pjy + pjz * pjz;

    // A operand: row M = j (= l16), K = v + 2h. h==0 -> (x,y); h==1 -> (z, ||p_j||^2)
    v2f A;
    A.x = h ? pjz  : pjx;
    A.y = h ? sq_j : pjy;

    v8f zero = {};
    // (neg_a, A, neg_b, B, c_mod, C, reuse_a, reuse_b)
    v8f Dg = __builtin_amdgcn_wmma_f32_16x16x4_f32(false, A, false, Bg,
                                                   (short)0, zero, false, false);
    v8f Dx = __builtin_amdgcn_wmma_f32_16x16x4_f32(false, A, false, Bx,
                                                   (short)0, zero, false, false);

    v8f res;
#pragma unroll
    for (int v = 0; v < 8; ++v) {
      const float along  = s_i - Dx[v];
      const float along2 = along * along;
      const float sq_dist = sq_i - 2.0f * Dg[v];
      float nsq = sq_dist + along2 * tn2m2;
      nsq = fmaxf(nsq, 0.0f);
      res[v] = alpha * nsq + beta * along2;
    }

    // 8 consecutive floats at out[i][j0 + 8h] -> two b128 stores (32B contiguous/lane)
    float* dst = outRow + (j0 + 8 * h);
    v4f lo = {res[0], res[1], res[2], res[3]};
    v4f hi = {res[4], res[5], res[6], res[7]};
    *(v4f*)(dst)     = lo;
    *(v4f*)(dst + 4) = hi;
  }
}

extern "C" void kernel_launch(void* const* d_in, const int* in_sizes, int n_in,
                              void* d_out, int out_size, void* d_ws, size_t ws_size,
                              hipStream_t stream) {
  const float* points = (const float*)d_in[0];
  const float* pdir   = (const float*)d_in[1];
  const float* lin    = (const float*)d_in[2];
  float* out = (float*)d_out;

  const int BN = in_sizes[2];            // linearity has B*N elements
  const int N  = (int)((long long)out_size / BN);  // out = B*N*N
  const int B  = BN / N;
  const int jBlocks = N / (16 * JT);     // j-blocks per row strip (N multiple of 256)

  const long long totalWaves = (long long)B * (N >> 4) * jBlocks;
  const int threadsPerBlock = 256;       // 8 waves (wave32)
  const long long totalThreads = totalWaves * 32;
  const int blocks = (int)((totalThreads + threadsPerBlock - 1) / threadsPerBlock);

  aniso_wmma_kernel<<<blocks, threadsPerBlock, 0, stream>>>(points, pdir, lin, out,
                                                            B, N, jBlocks);
}